// SinkhornDistance_7533372637840
// MI455X (gfx1250) — compile-verified
//
#include <hip/hip_runtime.h>
#include <hip/hip_bf16.h>
#include <math.h>

// ---------------------------------------------------------------------------
// Sinkhorn distance, MI455X (gfx1250, wave32).
//   P1 = P2 = 4096, D = 64, EPS = 0.1, MAX_ITER = 50, THRESH = 0.1
// Outputs (flat): [cost(1)] [pi(4096*4096)] [C(4096*4096)]
//
// gfx1250 paths used:
//  * V_WMMA_F32_16X16X4_F32 for the X·Y^T Gram matrix (only native f32 WMMA)
//  * global_load_async_to_lds_b32 + s_wait_asynccnt double-buffered staging of
//    C rows (+ the v vector) in the exp-bound logsumexp passes
// ---------------------------------------------------------------------------

#define P1 4096
#define P2 4096
#define DIM 64
#define EPSI 0.1f
#define INV_EPS 10.0f
#define MAX_ITER 50
#define THRESH 0.1f

typedef __attribute__((ext_vector_type(2))) float v2f;
typedef __attribute__((ext_vector_type(8))) float v8f;

#define S_WAIT_ASYNCCNT(n) asm volatile("s_wait_asynccnt %0" ::"i"(n))

// Stage one 4096-float row from global into LDS via the async-to-LDS engine.
// 256 threads, each lane moves 4B per issue; 16 issues cover 16 KB.
// ASYNCcnt += 16 per wave.  (b32 variant: C/pi live at a 4-byte-aligned offset
// inside d_out, so b64/b128 async ops would violate alignment rules.)
__device__ __forceinline__ void async_row_to_lds(float* dst_lds, const float* src, int t) {
  uint32_t l = (uint32_t)(uintptr_t)dst_lds + (uint32_t)t * 4u;
  uint64_t g = (uint64_t)(uintptr_t)(src + t);
#pragma unroll
  for (int c = 0; c < 16; ++c) {
    asm volatile("global_load_async_to_lds_b32 %0, %1, off"
                 ::"v"(l + (uint32_t)c * 1024u), "v"(g + (uint64_t)c * 4096ull)
                 : "memory");
  }
}

// ---------------------------------------------------------------------------
// Row norms ||x_i||^2 and ||y_j||^2 (tiny: 8192 rows of 64 floats)
// ---------------------------------------------------------------------------
__global__ void sk_norms(const float* __restrict__ x, const float* __restrict__ y,
                         float* __restrict__ xn, float* __restrict__ yn) {
  int gid = blockIdx.x * blockDim.x + threadIdx.x;       // 0..8191
  const float* src = (gid < P1) ? x : y;
  int row = (gid < P1) ? gid : (gid - P1);
  const float* p = src + row * DIM;
  float s = 0.f;
#pragma unroll
  for (int k = 0; k < DIM; k += 4) {
    float4 q = *(const float4*)(p + k);
    s += q.x * q.x + q.y * q.y + q.z * q.z + q.w * q.w;
  }
  if (gid < P1) xn[row] = s; else yn[row] = s;
}

// ---------------------------------------------------------------------------
// Cost matrix C = xn_i + yn_j - 2 * (X Y^T)  via V_WMMA_F32_16X16X4_F32.
// One wave32 per 16x16 output tile; K = 64 -> 16 WMMA ops per tile.
// A-fragment (16x4 f32): lanes 0-15 hold K=k0,k0+1; lanes 16-31 hold K=k0+2,3.
// C/D: VGPR r -> rows r (lanes 0-15) and r+8 (lanes 16-31).
// ---------------------------------------------------------------------------
__global__ void sk_cost_matrix(const float* __restrict__ x, const float* __restrict__ y,
                               const float* __restrict__ xn, const float* __restrict__ yn,
                               float* __restrict__ C) {
  const int wave   = threadIdx.x >> 5;                   // 4 waves / block
  const int tileId = blockIdx.x * 4 + wave;              // 0..65535
  const int tm     = tileId >> 8;                        // 0..255
  const int tn     = tileId & 255;                       // 0..255

  const int lane = threadIdx.x & 31;
  const int half = lane >> 4;                            // 0 | 1
  const int l15  = lane & 15;

  const float* xa = x + (tm * 16 + l15) * DIM + half * 2;
  const float* yb = y + (tn * 16 + l15) * DIM + half * 2;

  v8f acc = {};
#pragma unroll
  for (int k0 = 0; k0 < DIM; k0 += 4) {
    float2 af = *(const float2*)(xa + k0);
    float2 bf = *(const float2*)(yb + k0);
    v2f a = {af.x, af.y};
    v2f b = {bf.x, bf.y};
    acc = __builtin_amdgcn_wmma_f32_16x16x4_f32(false, a, false, b,
                                                (short)0, acc, false, false);
  }

  const int col = tn * 16 + l15;
  const float ync = yn[col];
#pragma unroll
  for (int r = 0; r < 8; ++r) {
    const int row = tm * 16 + r + half * 8;
    C[(long)row * P2 + col] = xn[row] + ync - 2.0f * acc[r];
  }
}

// ---------------------------------------------------------------------------
// Online (max,sum) combine for logsumexp
// ---------------------------------------------------------------------------
__device__ __forceinline__ void lse_accum(float a, float& m, float& s) {
  float nm = fmaxf(m, a);
  s = s * __expf(m - nm) + __expf(a - nm);
  m = nm;
}
__device__ __forceinline__ void lse_combine(float& m1, float& s1, float m2, float s2) {
  float M = fmaxf(m1, m2);
  s1 = s1 * __expf(m1 - M) + s2 * __expf(m2 - M);
  m1 = M;
}

// ---------------------------------------------------------------------------
// u-update: 4 rows per block (grid = 1024, 256 threads).
//   u_new_i = EPS*(log_mu - LSE_j((-C_ij + u_i + v_j)/EPS)) + u_i
// C rows and the v vector are DMA'd into LDS (double buffered) with
// global_load_async_to_lds_b32; compute overlaps the next row's transfer.
// ---------------------------------------------------------------------------
__global__ void sk_row_u(const float* __restrict__ C, float* __restrict__ u,
                         const float* __restrict__ v, float* __restrict__ du,
                         const int* __restrict__ done) {
  if (*done) return;
  __shared__ float buf[2][P2];     // 32 KB double buffer
  __shared__ float vbuf[P2];       // 16 KB, reused across the 4 rows
  __shared__ float sm[256], ss[256];
  const int t  = threadIdx.x;
  const int r0 = blockIdx.x * 4;

  async_row_to_lds(vbuf, v, t);                          // +16
  async_row_to_lds(buf[0], C + (long)r0 * P2, t);        // +16

#pragma unroll
  for (int r = 0; r < 4; ++r) {
    const int i = r0 + r;
    if (r < 3) {
      async_row_to_lds(buf[(r + 1) & 1], C + (long)(i + 1) * P2, t);  // +16
      S_WAIT_ASYNCCNT(16);   // in-order: v + row r resident, row r+1 in flight
    } else {
      S_WAIT_ASYNCCNT(0);
    }
    __syncthreads();

    const float ui = u[i];
    const float* row = buf[r & 1];
    float m = -INFINITY, s = 0.f;
#pragma unroll 4
    for (int j = t; j < P2; j += 256) {
      float a = (-row[j] + ui + vbuf[j]) * INV_EPS;
      lse_accum(a, m, s);
    }
    sm[t] = m; ss[t] = s;
    __syncthreads();
    for (int off = 128; off > 0; off >>= 1) {
      if (t < off) lse_combine(sm[t], ss[t], sm[t + off], ss[t + off]);
      __syncthreads();
    }
    if (t == 0) {
      const float log_mu = __logf(1.0f / (float)P1 + 1e-8f);
      float lse = sm[0] + __logf(ss[0]);
      float un = EPSI * (log_mu - lse) + ui;
      u[i] = un;
      du[i] = fabsf(un - ui);
    }
    __syncthreads();   // protect buf[r&1] before it is re-filled at r+2
  }
}

// ---------------------------------------------------------------------------
// v-update: one block (256 threads) per 64 columns; reduce over 4096 rows.
//   v_new_j = EPS*(log_nu - LSE_i((-C_ij + u_i + v_j)/EPS)) + v_j   (u is new)
// tcol = t&63 (column), trow = t>>6 -> 4 coalesced 256B row segments per step.
// ---------------------------------------------------------------------------
__global__ void sk_col_v(const float* __restrict__ C, const float* __restrict__ u,
                         float* __restrict__ v, const int* __restrict__ done) {
  if (*done) return;
  const int t    = threadIdx.x;
  const int tcol = t & 63;
  const int trow = t >> 6;                               // 0..3
  const int j    = blockIdx.x * 64 + tcol;
  const float vj = v[j];

  float m = -INFINITY, s = 0.f;
  for (int i = trow; i < P1; i += 4) {
    __builtin_prefetch(C + (long)(i + 8) * P2 + j, 0, 0);
    float a = (-C[(long)i * P2 + j] + u[i] + vj) * INV_EPS;
    lse_accum(a, m, s);
  }

  __shared__ float sm[256], ss[256];
  sm[t] = m; ss[t] = s;
  __syncthreads();
  if (trow == 0) {
    lse_combine(sm[tcol], ss[tcol], sm[tcol + 64],  ss[tcol + 64]);
    lse_combine(sm[tcol], ss[tcol], sm[tcol + 128], ss[tcol + 128]);
    lse_combine(sm[tcol], ss[tcol], sm[tcol + 192], ss[tcol + 192]);
    const float log_nu = __logf(1.0f / (float)P2 + 1e-8f);
    float lse = sm[tcol] + __logf(ss[tcol]);
    v[j] = EPSI * (log_nu - lse) + vj;
  }
}

// ---------------------------------------------------------------------------
// done |= (sum(du) < THRESH) — deterministic tree reduction, single block.
// ---------------------------------------------------------------------------
__global__ void sk_done_update(const float* __restrict__ du, int* __restrict__ done) {
  if (*done) return;
  const int t = threadIdx.x;
  float s = 0.f;
#pragma unroll
  for (int k = t; k < P1; k += 256) s += du[k];
  __shared__ float sh[256];
  sh[t] = s;
  __syncthreads();
  for (int off = 128; off > 0; off >>= 1) {
    if (t < off) sh[t] += sh[t + off];
    __syncthreads();
  }
  if (t == 0 && sh[0] < THRESH) *done = 1;
}

// ---------------------------------------------------------------------------
// pi = exp((-C + u_i + v_j)/EPS), per-row partials of sum(pi*C).
// 4 rows per block with the same async double-buffered staging.
// ---------------------------------------------------------------------------
__global__ void sk_finalize(const float* __restrict__ C, const float* __restrict__ u,
                            const float* __restrict__ v, float* __restrict__ pi,
                            float* __restrict__ part) {
  __shared__ float buf[2][P2];
  __shared__ float vbuf[P2];
  __shared__ float sh[256];
  const int t  = threadIdx.x;
  const int r0 = blockIdx.x * 4;

  async_row_to_lds(vbuf, v, t);
  async_row_to_lds(buf[0], C + (long)r0 * P2, t);

#pragma unroll
  for (int r = 0; r < 4; ++r) {
    const int i = r0 + r;
    if (r < 3) {
      async_row_to_lds(buf[(r + 1) & 1], C + (long)(i + 1) * P2, t);
      S_WAIT_ASYNCCNT(16);
    } else {
      S_WAIT_ASYNCCNT(0);
    }
    __syncthreads();

    const float ui = u[i];
    const float* row = buf[r & 1];
    float* prow = pi + (long)i * P2;
    float acc = 0.f;
#pragma unroll 4
    for (int j = t; j < P2; j += 256) {
      float c = row[j];
      float p = __expf((-c + ui + vbuf[j]) * INV_EPS);
      prow[j] = p;
      acc += p * c;
    }
    sh[t] = acc;
    __syncthreads();
    for (int off = 128; off > 0; off >>= 1) {
      if (t < off) sh[t] += sh[t + off];
      __syncthreads();
    }
    if (t == 0) part[i] = sh[0];
    __syncthreads();
  }
}

__global__ void sk_cost_reduce(const float* __restrict__ part, float* __restrict__ cost) {
  const int t = threadIdx.x;
  float s = 0.f;
#pragma unroll
  for (int k = t; k < P1; k += 256) s += part[k];
  __shared__ float sh[256];
  sh[t] = s;
  __syncthreads();
  for (int off = 128; off > 0; off >>= 1) {
    if (t < off) sh[t] += sh[t + off];
    __syncthreads();
  }
  if (t == 0) cost[0] = sh[0];
}

// ---------------------------------------------------------------------------
// Launch
// ---------------------------------------------------------------------------
extern "C" void kernel_launch(void* const* d_in, const int* in_sizes, int n_in,
                              void* d_out, int out_size, void* d_ws, size_t ws_size,
                              hipStream_t stream) {
  (void)in_sizes; (void)n_in; (void)out_size; (void)ws_size;

  const float* x = (const float*)d_in[0];
  const float* y = (const float*)d_in[1];

  float* out  = (float*)d_out;
  float* cost = out;                                   // [1]
  float* pi   = out + 1;                               // [4096*4096]
  float* Cm   = out + 1 + (long)P1 * P2;               // [4096*4096]

  float* ws   = (float*)d_ws;
  float* u    = ws;                                    // 4096
  float* v    = ws + 4096;                             // 4096
  float* du   = ws + 8192;                             // 4096
  int*  done  = (int*)(ws + 12288);                    // 1
  float* xn   = ws + 12352;                            // 4096
  float* yn   = ws + 16448;                            // 4096
  float* part = ws + 20544;                            // 4096

  // re-init mutable state every call (u, v, du, done)
  hipMemsetAsync(d_ws, 0, (size_t)12352 * sizeof(float), stream);

  sk_norms<<<(2 * P1) / 256, 256, 0, stream>>>(x, y, xn, yn);
  sk_cost_matrix<<<(P1 / 16) * (P2 / 16) / 4, 128, 0, stream>>>(x, y, xn, yn, Cm);

  for (int it = 0; it < MAX_ITER; ++it) {
    sk_row_u<<<P1 / 4, 256, 0, stream>>>(Cm, u, v, du, done);
    sk_col_v<<<P2 / 64, 256, 0, stream>>>(Cm, u, v, done);
    sk_done_update<<<1, 256, 0, stream>>>(du, done);
  }

  sk_finalize<<<P1 / 4, 256, 0, stream>>>(Cm, u, v, pi, part);
  sk_cost_reduce<<<1, 256, 0, stream>>>(part, cost);
}